// Apply2DDispField_5506148074184
// MI455X (gfx1250) — compile-verified
//
#include <hip/hip_runtime.h>

// Apply2DDispField: bilinear warp with rounded corners, B=16, H=W=1024, C=1.
// Bandwidth-bound gather. Strategy for MI455X (23.3 TB/s HBM, 192 MB L2):
//  - DispField (128 MB) and output (64 MB) are touched exactly once -> stream
//    them with non-temporal 128-bit loads/stores so they don't evict the image
//    from L2.
//  - Img (64 MB total) fits entirely in the 192 MB L2 -> gather with default
//    (regular-temporal) loads so the random gathers run at L2 bandwidth.
//  - Branchless gathers: wrap the padded index (1024 -> 0) so every load is
//    in-bounds and unconditional, then zero the padded samples with a single
//    select. No exec-mask divergence, loads stay clause-able.
//  - 4 pixels per thread, wave32-safe (no lane-count assumptions).

#define B_  16
#define H_  1024
#define W_  1024

typedef float float4v __attribute__((ext_vector_type(4)));

__global__ __launch_bounds__(256) void
apply_dispfield_kernel(const float* __restrict__ img,
                       const float* __restrict__ disp,
                       float* __restrict__ out)
{
    const int tid = blockIdx.x * blockDim.x + threadIdx.x;
    const long long p0 = (long long)tid * 4;          // first of 4 consecutive pixels
    // flat pixel index = ((b*H + i)*W + j); W divisible by 4 so all 4 share (b,i)
    const int j0  = (int)(p0 & (W_ - 1));
    const int ihw = (int)(p0 >> 10);                  // b*H + i
    const int i   = ihw & (H_ - 1);
    const int b   = ihw >> 10;

    // Stream displacement: 8 consecutive floats = two 128-bit NT loads
    const float* dptr = disp + p0 * 2;
    const float4v d01 = __builtin_nontemporal_load((const float4v*)(dptr));
    const float4v d23 = __builtin_nontemporal_load((const float4v*)(dptr + 4));

    const float* __restrict__ imgb = img + (size_t)b * (H_ * W_);

    // identity grid: x along height, y along width, linspace(-1,1,1024)
    const float step = 2.0f / 1023.0f;
    const float xt = (float)i * step - 1.0f;          // same for all 4 pixels

    float dxv[4] = { d01.x, d01.z, d23.x, d23.z };
    float dyv[4] = { d01.y, d01.w, d23.y, d23.w };

    float4v res;
#pragma unroll
    for (int k = 0; k < 4; ++k) {
        const float yt = (float)(j0 + k) * step - 1.0f;
        // unnormalize: 0.5*(grid - disp + 1) * (dim - 1); ((H+1)-1) - 1 = 1023
        const float x = 0.5f * (xt - dxv[k] + 1.0f) * 1023.0f;
        const float y = 0.5f * (yt - dyv[k] + 1.0f) * 1023.0f;

        // jnp.round = round half to even -> v_rndne / v_cvt_i32_f32 (RNE)
        const int rx = __float2int_rn(x);
        const int ry = __float2int_rn(y);
        const int x0 = min(max(rx,     0), H_);       // clip to [0, 1024]
        const int x1 = min(max(rx + 1, 0), H_);
        const int y0 = min(max(ry,     0), W_);
        const int y1 = min(max(ry + 1, 0), W_);

        // Index 1024 is the zero-pad row/col. Wrap it to 0 so the load is
        // always in-bounds (unconditional, clause-friendly), then zero the
        // sample with one select. valid iff ((xi|yi) & 1024) == 0.
        const float v00 = imgb[((x0 & 1023) << 10) + (y0 & 1023)];
        const float v01 = imgb[((x0 & 1023) << 10) + (y1 & 1023)];
        const float v10 = imgb[((x1 & 1023) << 10) + (y0 & 1023)];
        const float v11 = imgb[((x1 & 1023) << 10) + (y1 & 1023)];
        const float I00 = ((x0 | y0) & 1024) ? 0.0f : v00;
        const float I01 = ((x0 | y1) & 1024) ? 0.0f : v01;
        const float I10 = ((x1 | y0) & 1024) ? 0.0f : v10;
        const float I11 = ((x1 | y1) & 1024) ? 0.0f : v11;

        const float x0f = (float)x0, x1f = (float)x1;
        const float y0f = (float)y0, y1f = (float)y1;

        const float w00 = (x1f - x) * (y1f - y);
        const float w01 = (x1f - x) * (y - y0f);
        const float w10 = (x - x0f) * (y1f - y);
        const float w11 = (x - x0f) * (y - y0f);

        float r = w00 * I00;
        r = fmaf(w01, I01, r);
        r = fmaf(w10, I10, r);
        r = fmaf(w11, I11, r);
        res[k] = r;
    }

    // output written once -> non-temporal 128-bit store
    __builtin_nontemporal_store(res, (float4v*)(out + p0));
}

extern "C" void kernel_launch(void* const* d_in, const int* in_sizes, int n_in,
                              void* d_out, int out_size, void* d_ws, size_t ws_size,
                              hipStream_t stream) {
    (void)in_sizes; (void)n_in; (void)d_ws; (void)ws_size; (void)out_size;
    const float* img  = (const float*)d_in[0];   // (B,H,W,1) float32
    const float* disp = (const float*)d_in[1];   // (B,H,W,2) float32
    float* out        = (float*)d_out;           // (B,H,W,1) float32

    const long long npix = (long long)B_ * H_ * W_;  // 16,777,216
    const int nthreads   = (int)(npix / 4);          // 4 pixels per thread
    const int block      = 256;                      // 8 wave32 per block
    const int grid       = nthreads / block;         // 16384 blocks

    apply_dispfield_kernel<<<grid, block, 0, stream>>>(img, disp, out);
}